// IMPALA_49941879718454
// MI455X (gfx1250) — compile-verified
//
#include <hip/hip_runtime.h>
#include <hip/hip_bf16.h>
#include <stdint.h>

// IMPALA / V-trace loss for MI455X (gfx1250, wave32).
//  T=256, B=2048, A=18. Memory-bound: policy read once (37.7MB, coalesced
//  row reads), behavior gathered, [T,B] arrays streamed. ~45-75MB -> ~3us
//  at 23.3 TB/s; everything is L2-resident (192MB) after first touch.
//  All mean-reductions use V_WMMA_F32_16X16X4_F32 as a 32-lane adder tree.

#define T_DIM   256
#define B_DIM   2048
#define A_DIM   18
#define N_TB    (T_DIM * B_DIM)          // 524288
#define GAMMA_F 0.99f
#define ENT_COEF 0.01f

typedef __attribute__((ext_vector_type(2))) float v2f;
typedef __attribute__((ext_vector_type(8))) float v8f;

// ---------------------------------------------------------------------------
// Wave32 sum-reduction via V_WMMA_F32_16X16X4_F32.
// A (16x4 f32): lane L<16 comp0 -> A[L][0]; lane L>=16 comp0 -> A[L-16][2].
// With B = ones(4x16):  D[m][n] = x_m + x_{m+16}  (identical for every n).
// D layout: lanes 0-15 VGPR r = D[r][lane]; lanes 16-31 VGPR r = D[8+r][lane-16].
// Sum of the 8 D regs gives rows 0-7 (lanes<16) / rows 8-15 (lanes>=16);
// one ds_swizzle SWAPX16 (xor 0x10 -> offset 0x401F) completes the total.
// NOTE: WMMA requires EXEC all-ones -> only call from reconverged code.
// ---------------------------------------------------------------------------
__device__ __forceinline__ float wave_reduce_f32_wmma(float x) {
  v2f a; a[0] = x;    a[1] = 0.0f;
  v2f b; b[0] = 1.0f; b[1] = 1.0f;
  v8f c = {};
  c = __builtin_amdgcn_wmma_f32_16x16x4_f32(
      /*neg_a=*/false, a, /*neg_b=*/false, b,
      /*c_mod=*/(short)0, c, /*reuse_a=*/false, /*reuse_b=*/false);
  float r = c[0] + c[1] + c[2] + c[3] + c[4] + c[5] + c[6] + c[7];
  // swap halves: group-of-32 swizzle, xor_mask=0x10, and_mask=0x1f
  int ri = __builtin_amdgcn_ds_swizzle(__float_as_int(r), 0x401F);
  return r + __int_as_float(ri);
}

__device__ __forceinline__ float block_reduce_f32(float x, float* smem) {
  float w = wave_reduce_f32_wmma(x);
  const int lane = threadIdx.x & 31;
  const int wid  = threadIdx.x >> 5;
  if (lane == 0) smem[wid] = w;
  __syncthreads();
  float total = 0.0f;
  if (wid == 0) {  // wave-uniform branch: wave 0 keeps full EXEC inside
    const int nwaves = (int)(blockDim.x >> 5);
    float v = (lane < nwaves) ? smem[lane] : 0.0f;
    total = wave_reduce_f32_wmma(v);
  }
  __syncthreads();   // make smem reusable by a subsequent call
  return total;      // valid in all lanes of wave 0
}

// ---------------------------------------------------------------------------
// Kernel 1 (fused prep + entropy): one thread per (t,b).
//  * reads its contiguous 18-float policy row (9 x b64, wave reads 2304
//    contiguous bytes -> line-coalesced), accumulating p*log(p+1e-10) and
//    selecting p[action] from registers (no second gather into policy),
//  * gathers behavior[t,b,a] (the only remaining gather),
//  * emits two float4 workspace arrays consumed by the scan:
//      g1 = {delta, disc*c, disc, pg_adv_vt}, g2 = {logp, clipped_r, v, 0}
//  * WMMA block-reduces the entropy partial -> one atomic per block.
// ---------------------------------------------------------------------------
__global__ void impala_prep_entropy_kernel(
    const float* __restrict__ policy, const float* __restrict__ behavior,
    const float* __restrict__ rewards, const float* __restrict__ values,
    const float* __restrict__ next_values, const int* __restrict__ actions,
    const uint8_t* __restrict__ done,
    float4* __restrict__ g1, float4* __restrict__ g2,
    float* __restrict__ acc) {
  const int idx = blockIdx.x * blockDim.x + threadIdx.x;   // < N_TB (exact grid)
  const int t   = idx / B_DIM;                             // uniform per block

  const int a = actions[idx];
  const float2* __restrict__ row =
      (const float2*)(policy + (size_t)idx * A_DIM);       // 8-byte aligned

  float ent = 0.0f;
  float p   = 0.0f;
  #pragma unroll
  for (int k = 0; k < A_DIM / 2; ++k) {
    const float2 c2 = row[k];
    ent += c2.x * __logf(c2.x + 1e-10f);
    ent += c2.y * __logf(c2.y + 1e-10f);
    p = (a == 2 * k)     ? c2.x : p;
    p = (a == 2 * k + 1) ? c2.y : p;
  }

  const float mu   = behavior[(size_t)idx * A_DIM + (size_t)a];
  const float lp   = __logf(p);
  const float rho  = __expf(lp - __logf(mu));      // == p/mu
  const float crho = fminf(1.0f, rho);             // rho clip & cs (thresh 1.0)

  const float cr = fminf(1.0f, fmaxf(-1.0f, rewards[idx]));
  const float dc = done[idx] ? 0.0f : GAMMA_F;     // (~done)*gamma
  const float v  = values[idx];
  const float nv = next_values[idx];
  float v1 = nv;                                   // bootstrap at t = T-1
  if (t < T_DIM - 1) v1 = values[idx + B_DIM];     // values[t+1,b] (uniform br.)

  g1[idx] = make_float4(crho * fmaf(dc, v1, cr - v),  // delta
                        dc * crho,                    // scan coefficient
                        dc,                           // discount
                        crho * fmaf(dc, nv, cr - v)); // pg_adv_vt
  g2[idx] = make_float4(lp, cr, v, 0.0f);

  __shared__ float smem[8];
  const float tot = block_reduce_f32(ent, smem);
  if (threadIdx.x == 0) atomicAdd(&acc[0], tot);
}

// ---------------------------------------------------------------------------
// Kernel 2: reverse scan over T per column b (2 x b128 per step, coalesced
// across b) fused with policy/value loss partials.
//   vs_mv[t] = delta[t] + (disc*c)[t] * vs_mv[t+1]
//   value term:  (vs - values)^2 == vs_mv^2
//   policy term: pg_adv_vt * (cr + disc*vs - v) * logp
// ---------------------------------------------------------------------------
__global__ void impala_scan_loss_kernel(
    const float4* __restrict__ g1, const float4* __restrict__ g2,
    float* __restrict__ acc) {
  const int b = blockIdx.x * blockDim.x + threadIdx.x;   // 0 .. B_DIM-1

  float vsmv  = 0.0f;   // running vs_minus_v_xs (reverse scan accumulator)
  float pol_s = 0.0f;
  float val_s = 0.0f;
  for (int t = T_DIM - 1; t >= 0; --t) {
    const int i = t * B_DIM + b;
    const float4 q1 = g1[i];                 // {delta, disc*c, disc, padv}
    const float4 q2 = g2[i];                 // {logp, cr, v, -}
    vsmv = fmaf(q1.y, vsmv, q1.x);           // scan step
    const float vs  = vsmv + q2.z;
    const float adv = q1.w * fmaf(q1.z, vs, q2.y - q2.z);  // pg_advantage
    pol_s += adv * q2.x;
    val_s += vsmv * vsmv;
  }

  __shared__ float smem[8];
  const float pol_t = block_reduce_f32(pol_s, smem);
  const float val_t = block_reduce_f32(val_s, smem);
  if (threadIdx.x == 0) {
    atomicAdd(&acc[1], pol_t);
    atomicAdd(&acc[2], val_t);
  }
}

// ---------------------------------------------------------------------------
// Kernel 3: combine sums into the scalar loss.
//   loss = (val_sum - pol_sum - COEF*ent_sum) / (T*B)
// ---------------------------------------------------------------------------
__global__ void impala_finalize_kernel(const float* __restrict__ acc,
                                       float* __restrict__ out) {
  if (blockIdx.x == 0 && threadIdx.x == 0) {
    const float ent_s = acc[0];   // sum p*log(p+eps)   -> entropy_loss = -ent/N
    const float pol_s = acc[1];   // sum adv*logp       -> policy_loss  = -pol/N
    const float val_s = acc[2];   // sum vs_mv^2        -> value_loss   =  val/N
    const float invN  = 1.0f / (float)N_TB;
    out[0] = (val_s - pol_s - ENT_COEF * ent_s) * invN;
  }
}

extern "C" void kernel_launch(void* const* d_in, const int* in_sizes, int n_in,
                              void* d_out, int out_size, void* d_ws, size_t ws_size,
                              hipStream_t stream) {
  const float*   policy   = (const float*)d_in[0];
  const float*   behavior = (const float*)d_in[1];
  const float*   rewards  = (const float*)d_in[2];
  const float*   values   = (const float*)d_in[3];
  const float*   next_v   = (const float*)d_in[4];
  const int*     actions  = (const int*)d_in[5];
  const uint8_t* done     = (const uint8_t*)d_in[6];   // jax bool = 1 byte/elem

  // Workspace layout: 2 x float4[T*B] (16 MB) + 3 float accumulators.
  float4* g1  = (float4*)d_ws;
  float4* g2  = g1 + (size_t)N_TB;
  float*  acc = (float*)(g2 + (size_t)N_TB);   // [ent_sum, pol_sum, val_sum]

  hipMemsetAsync(acc, 0, 3 * sizeof(float), stream);   // graph-capture safe

  impala_prep_entropy_kernel<<<N_TB / 256, 256, 0, stream>>>(
      policy, behavior, rewards, values, next_v, actions, done, g1, g2, acc);

  impala_scan_loss_kernel<<<B_DIM / 256, 256, 0, stream>>>(g1, g2, acc);

  impala_finalize_kernel<<<1, 64, 0, stream>>>(acc, (float*)d_out);
}